// soft_cldice_35055523070188
// MI455X (gfx1250) — compile-verified
//
#include <hip/hip_runtime.h>
#include <stdint.h>

#ifndef __has_builtin
#define __has_builtin(x) 0
#endif

#define TILE 64
#define AP   72     // LDS row pitch (words); 288B rows keep b128 stores aligned
#define DIMH 512
#define DIMW 512

typedef int v4i __attribute__((ext_vector_type(4)));
typedef __attribute__((address_space(1))) int  gint_t;    // global int
typedef __attribute__((address_space(3))) int  lint_t;    // LDS int
typedef __attribute__((address_space(1))) v4i  gv4i_t;    // global int4
typedef __attribute__((address_space(3))) v4i  lv4i_t;    // LDS int4

// ---- CDNA5 async global->LDS copies (ASYNCcnt path) -------------------------
__device__ __forceinline__ void async_ld_f32(const float* g, float* l) {
#if __has_builtin(__builtin_amdgcn_global_load_async_to_lds_b32)
  __builtin_amdgcn_global_load_async_to_lds_b32((gint_t*)g, (lint_t*)l, 0, 0);
#elif defined(__gfx1250__)
  unsigned lofs = (unsigned)(uintptr_t)l;
  asm volatile("global_load_async_to_lds_b32 %0, %1, off"
               :: "v"(lofs), "v"(g) : "memory");
#else
  *l = *g;
#endif
}

__device__ __forceinline__ void async_ld_f32x4(const float* g, float* l) {
#if __has_builtin(__builtin_amdgcn_global_load_async_to_lds_b128)
  __builtin_amdgcn_global_load_async_to_lds_b128((gv4i_t*)g, (lv4i_t*)l, 0, 0);
#else
  async_ld_f32(g + 0, l + 0);
  async_ld_f32(g + 1, l + 1);
  async_ld_f32(g + 2, l + 2);
  async_ld_f32(g + 3, l + 3);
#endif
}

__device__ __forceinline__ void wait_async0() {
#if __has_builtin(__builtin_amdgcn_s_wait_asynccnt)
  __builtin_amdgcn_s_wait_asynccnt(0);
#elif defined(__gfx1250__)
  asm volatile("s_wait_asynccnt 0" ::: "memory");
#endif
}

// ---- one fused soft_skel iteration ------------------------------------------
// A columns map gj = j0-4+lj (lj in [0,72)); rows map gi = i0-3+li (li in [0,70)).
// MODE 0: skel = relu(img - dilate(erode(img))); imgOut = img        (init)
// MODE 1: img' = erode(img); d = relu(img' - dilate(erode(img')));
//         skel += relu(d - skel*d); imgOut = img'                    (step)
template <int MODE>
__global__ __launch_bounds__(256) void skel_step_kernel(
    const float* __restrict__ imgIn, float* __restrict__ imgOut,
    float* __restrict__ skel)
{
  __shared__ float A [70 * AP];   // img tile + halo
  __shared__ float E1[68 * AP];   // erode(img)
  __shared__ float E2[66 * AP];   // erode(erode(img))

  constexpr int H = DIMH, W = DIMW;
  constexpr int tilesX = W / TILE;            // 8
  constexpr int tpp    = tilesX * (H / TILE); // 64  (power of two -> shifts)
  const int tid   = threadIdx.x;
  const int plane = blockIdx.x / tpp;
  const int t     = blockIdx.x - plane * tpp;
  const int i0    = (t / tilesX) * TILE;
  const int j0    = (t % tilesX) * TILE;
  const size_t pOff = (size_t)plane * H * W;
  const float PINF  = __builtin_inff();

  // ---- stage halo tile into LDS via async DMA ----
  const bool interior = (i0 >= TILE) && (i0 + TILE < H) &&
                        (j0 >= TILE) && (j0 + TILE < W);
  if (interior) {
    // 70 rows x 18 b128 chunks, fully in-bounds & 16B-aligned on both sides
    for (int k = tid; k < 70 * 18; k += 256) {
      int li = k / 18, cj = (k - li * 18) * 4;
      async_ld_f32x4(imgIn + pOff + (size_t)(i0 - 3 + li) * W + (j0 - 4 + cj),
                     &A[li * AP + cj]);
    }
  } else {
    for (int k = tid; k < 70 * AP; k += 256) {
      int li = k / AP, lj = k - li * AP;
      int gi = i0 - 3 + li, gj = j0 - 4 + lj;
      float* dst = &A[li * AP + lj];
      if ((unsigned)gi < (unsigned)H && (unsigned)gj < (unsigned)W)
        async_ld_f32(imgIn + pOff + (size_t)gi * W + gj, dst);
      else
        *dst = PINF;   // erode-neutral
    }
  }
  wait_async0();
  __syncthreads();

  // ---- E1 = erode(A) on 68x68; center (mi,mj) <-> global (i0-2+mi, j0-2+mj)
  // OOB centers: +inf if consumed by another erode (MODE 1),
  //              -inf if consumed directly by dilate (MODE 0).
  constexpr float FILL_E1 = (MODE == 0) ? -__builtin_inff() : __builtin_inff();
  for (int k = tid; k < 68 * 68; k += 256) {
    int mi = k / 68, mj = k - mi * 68;
    int gi = i0 - 2 + mi, gj = j0 - 2 + mj;
    float v;
    if ((unsigned)gi < (unsigned)H && (unsigned)gj < (unsigned)W) {
      int c = (mi + 1) * AP + (mj + 2);
      float ctr  = A[c];
      float vmin = fminf(fminf(A[c - AP], ctr), A[c + AP]);
      float hmin = fminf(fminf(A[c - 1],  ctr), A[c + 1]);
      v = fminf(vmin, hmin);
    } else {
      v = FILL_E1;
    }
    E1[mi * AP + mj] = v;
  }
  __syncthreads();

  if (MODE != 0) {
    // ---- E2 = erode(E1) on 66x66; center <-> global (i0-1+ni, j0-1+nj)
    for (int k = tid; k < 66 * 66; k += 256) {
      int ni = k / 66, nj = k - ni * 66;
      int gi = i0 - 1 + ni, gj = j0 - 1 + nj;
      float v;
      if ((unsigned)gi < (unsigned)H && (unsigned)gj < (unsigned)W) {
        int c = (ni + 1) * AP + (nj + 1);
        float ctr  = E1[c];
        float vmin = fminf(fminf(E1[c - AP], ctr), E1[c + AP]);
        float hmin = fminf(fminf(E1[c - 1],  ctr), E1[c + 1]);
        v = fminf(vmin, hmin);
      } else {
        v = -PINF;   // dilate-neutral
      }
      E2[ni * AP + nj] = v;
    }
    __syncthreads();
  }

  // ---- dilate 3x3 + skeleton update on the 64x64 core ----
  for (int k = tid; k < TILE * TILE; k += 256) {
    int oi = k / TILE, oj = k - oi * TILE;
    size_t g = pOff + (size_t)(i0 + oi) * W + (j0 + oj);
    if (MODE == 0) {
      int c = (oi + 2) * AP + (oj + 2);
      float m0 = fmaxf(fmaxf(E1[c - AP - 1], E1[c - AP]), E1[c - AP + 1]);
      float m1 = fmaxf(fmaxf(E1[c - 1],      E1[c]),      E1[c + 1]);
      float m2 = fmaxf(fmaxf(E1[c + AP - 1], E1[c + AP]), E1[c + AP + 1]);
      float open_ = fmaxf(fmaxf(m0, m1), m2);
      float img   = A[(oi + 3) * AP + (oj + 4)];
      skel[g]   = fmaxf(img - open_, 0.0f);
      imgOut[g] = img;
    } else {
      int c = (oi + 1) * AP + (oj + 1);
      float m0 = fmaxf(fmaxf(E2[c - AP - 1], E2[c - AP]), E2[c - AP + 1]);
      float m1 = fmaxf(fmaxf(E2[c - 1],      E2[c]),      E2[c + 1]);
      float m2 = fmaxf(fmaxf(E2[c + AP - 1], E2[c + AP]), E2[c + AP + 1]);
      float open_  = fmaxf(fmaxf(m0, m1), m2);
      float imgNew = E1[(oi + 2) * AP + (oj + 2)];
      float delta  = fmaxf(imgNew - open_, 0.0f);
      float s = skel[g];
      skel[g]   = s + fmaxf(delta - s * delta, 0.0f);
      imgOut[g] = imgNew;
    }
  }
}

// ---- channel-1 reductions: acc[0]+=sum(skel), acc[1]+=sum(skel*other) -------
__global__ __launch_bounds__(256) void reduce_kernel(
    const float* __restrict__ skel, const float* __restrict__ other,
    float* __restrict__ acc, int N, int C)
{
  __shared__ float s0[256];
  __shared__ float s1[256];
  constexpr long hw = (long)DIMH * DIMW;
  const long per_n  = (long)(C - 1) * hw;
  const long total  = (long)N * per_n;
  float a0 = 0.0f, a1 = 0.0f;
  for (long idx = (long)blockIdx.x * 256 + threadIdx.x; idx < total;
       idx += (long)gridDim.x * 256) {
    long n = idx / per_n;
    long r = idx - n * per_n;
    long c = 1 + r / hw;
    long p = r - (c - 1) * hw;
    long off = (n * C + c) * hw + p;
    float sv = skel[off];
    a0 += sv;
    a1 += sv * other[off];
  }
  s0[threadIdx.x] = a0; s1[threadIdx.x] = a1;
  __syncthreads();
  for (int st = 128; st > 0; st >>= 1) {
    if (threadIdx.x < st) {
      s0[threadIdx.x] += s0[threadIdx.x + st];
      s1[threadIdx.x] += s1[threadIdx.x + st];
    }
    __syncthreads();
  }
  if (threadIdx.x == 0) {
    atomicAdd(&acc[0], s0[0]);
    atomicAdd(&acc[1], s1[0]);
  }
}

__global__ void zero4_kernel(float* acc) {
  if (threadIdx.x < 4) acc[threadIdx.x] = 0.0f;
}

__global__ void finalize_kernel(const float* __restrict__ acc, float* __restrict__ out) {
  if (threadIdx.x == 0 && blockIdx.x == 0) {
    float sp  = acc[0], spw = acc[1];   // sum(skel_pred), sum(skel_pred*y_true)
    float st  = acc[2], stw = acc[3];   // sum(skel_true), sum(skel_true*y_pred)
    float tprec = (spw + 1.0f) / (sp + 1.0f);
    float tsens = (stw + 1.0f) / (st + 1.0f);
    out[0] = 1.0f - 2.0f * (tprec * tsens) / (tprec + tsens);
  }
}

extern "C" void kernel_launch(void* const* d_in, const int* in_sizes, int n_in,
                              void* d_out, int out_size, void* d_ws, size_t ws_size,
                              hipStream_t stream) {
  (void)n_in; (void)out_size; (void)ws_size;
  const float* y_true = (const float*)d_in[0];
  const float* y_pred = (const float*)d_in[1];

  const int HW = DIMH * DIMW;
  const int planes = in_sizes[0] / HW;      // N*C = 16
  const int C = 2;
  const int N = planes / C;
  const size_t tElems = (size_t)planes * HW;

  float* imgA = (float*)d_ws;
  float* imgB = imgA + tElems;
  float* skel = imgB + tElems;
  float* acc  = skel + tElems;              // 4 floats

  dim3 blk(256);
  dim3 grd((unsigned)(planes * (DIMH / TILE) * (DIMW / TILE)));

  zero4_kernel<<<1, 32, 0, stream>>>(acc);

  // ---------------- pred pass ----------------
  skel_step_kernel<0><<<grd, blk, 0, stream>>>(y_pred, imgA, skel);
  {
    const float* cur = imgA; float* nxt = imgB;
    for (int it = 0; it < 100; ++it) {
      skel_step_kernel<1><<<grd, blk, 0, stream>>>(cur, nxt, skel);
      float* tmp = (float*)cur; cur = nxt; nxt = tmp;
    }
  }
  reduce_kernel<<<1024, blk, 0, stream>>>(skel, y_true, acc + 0, N, C);

  // ---------------- true pass ----------------
  skel_step_kernel<0><<<grd, blk, 0, stream>>>(y_true, imgA, skel);
  {
    const float* cur = imgA; float* nxt = imgB;
    for (int it = 0; it < 100; ++it) {
      skel_step_kernel<1><<<grd, blk, 0, stream>>>(cur, nxt, skel);
      float* tmp = (float*)cur; cur = nxt; nxt = tmp;
    }
  }
  reduce_kernel<<<1024, blk, 0, stream>>>(skel, y_pred, acc + 2, N, C);

  finalize_kernel<<<1, 1, 0, stream>>>(acc, (float*)d_out);
}